// SBMemoryWriter_28587302323143
// MI455X (gfx1250) — compile-verified
//
#include <hip/hip_runtime.h>
#include <math.h>

typedef __attribute__((ext_vector_type(2))) float v2f;
typedef __attribute__((ext_vector_type(8))) float v8f;

#define B_ 1024
#define N_ 128
#define D_ 512

__device__ __forceinline__ float sigm(float x) { return 1.0f / (1.0f + expf(-x)); }
__device__ __forceinline__ float clamp01(float x) { return fminf(fmaxf(x, 0.0f), 1.0f); }

// ---------------------------------------------------------------------------
// K0: per-batch-row D->1 projections: write_strength, importance, mg_lin, bd_lin
// ---------------------------------------------------------------------------
__global__ __launch_bounds__(256) void k_rowproj(
    const float* __restrict__ H,
    const float* __restrict__ Wwr, const float* __restrict__ bwr,
    const float* __restrict__ Wim, const float* __restrict__ bim,
    const float* __restrict__ Wmg, const float* __restrict__ bmg,
    const float* __restrict__ Wbd, const float* __restrict__ bbd,
    float* __restrict__ wstr_ws, float* __restrict__ out_wstr,
    float* __restrict__ imp_ws, float* __restrict__ mg_ws, float* __restrict__ bd_ws)
{
    const int b = blockIdx.x;
    const int t = threadIdx.x;
    float s0 = 0.f, s1 = 0.f, s2 = 0.f, s3 = 0.f;
    for (int i = t; i < D_; i += 256) {
        const float h = H[b * D_ + i];
        s0 = fmaf(h, Wwr[i], s0);
        s1 = fmaf(h, Wim[i], s1);
        s2 = fmaf(h, Wmg[i], s2);
        s3 = fmaf(h, Wbd[i], s3);
    }
    for (int off = 16; off; off >>= 1) {
        s0 += __shfl_xor(s0, off);
        s1 += __shfl_xor(s1, off);
        s2 += __shfl_xor(s2, off);
        s3 += __shfl_xor(s3, off);
    }
    __shared__ float red[8][4];
    const int lane = t & 31, w = t >> 5;
    if (lane == 0) { red[w][0] = s0; red[w][1] = s1; red[w][2] = s2; red[w][3] = s3; }
    __syncthreads();
    if (t == 0) {
        float a0 = 0.f, a1 = 0.f, a2 = 0.f, a3 = 0.f;
        for (int i = 0; i < 8; ++i) { a0 += red[i][0]; a1 += red[i][1]; a2 += red[i][2]; a3 += red[i][3]; }
        const float ws = sigm(a0 + bwr[0]);
        wstr_ws[b] = ws;
        out_wstr[b] = ws;
        imp_ws[b] = sigm(a1 + bim[0]);
        mg_ws[b] = a2 + bmg[0];
        bd_ws[b] = a3 + bbd[0];
    }
}

// ---------------------------------------------------------------------------
// K1: candidate = tanh(H @ W + b) via V_WMMA_F32_16X16X4_F32
// grid = (B/16, (D/16)/8, 2), block = 256 (8 waves, one 16x16 C tile each)
// ---------------------------------------------------------------------------
__global__ __launch_bounds__(256) void k_gemm_tanh(
    const float* __restrict__ H,
    const float* __restrict__ Wk, const float* __restrict__ bk,
    const float* __restrict__ Wv, const float* __restrict__ bv,
    float* __restrict__ ck, float* __restrict__ cv)
{
    const int lane = threadIdx.x & 31;
    const int wave = threadIdx.x >> 5;
    const int m0 = blockIdx.x * 16;
    const int n0 = (blockIdx.y * 8 + wave) * 16;
    const float* __restrict__ W  = blockIdx.z ? Wv : Wk;
    const float* __restrict__ bb = blockIdx.z ? bv : bk;
    float* __restrict__ out      = blockIdx.z ? cv : ck;

    const int mrow  = m0 + (lane & 15);
    const int ncol  = n0 + (lane & 15);
    const int khalf = (lane >> 4) * 2;   // lanes 0-15 -> K 0,1 ; lanes 16-31 -> K 2,3

    v8f c = {};
    for (int k0 = 0; k0 < D_; k0 += 4) {
        const int k = k0 + khalf;
        v2f a, bf;
        a.x  = H[mrow * D_ + k];
        a.y  = H[mrow * D_ + k + 1];
        bf.x = W[k * D_ + ncol];
        bf.y = W[(k + 1) * D_ + ncol];
        c = __builtin_amdgcn_wmma_f32_16x16x4_f32(false, a, false, bf, (short)0, c, false, false);
    }
    const float bn = bb[ncol];
    const int mbase = m0 + (lane >> 4) * 8;  // C: vgpr r holds M=r (lanes<16) / M=r+8
    #pragma unroll
    for (int r = 0; r < 8; ++r)
        out[(mbase + r) * D_ + ncol] = tanhf(c[r] + bn);
}

// ---------------------------------------------------------------------------
// K2: cknorm[b] = max(||candidate_key[b]||, 1e-6)
// ---------------------------------------------------------------------------
__global__ __launch_bounds__(256) void k_cknorm(const float* __restrict__ ck,
                                                float* __restrict__ cknorm)
{
    const int b = blockIdx.x, t = threadIdx.x;
    float s = 0.f;
    for (int i = t; i < D_; i += 256) { const float v = ck[b * D_ + i]; s = fmaf(v, v, s); }
    for (int off = 16; off; off >>= 1) s += __shfl_xor(s, off);
    __shared__ float red[8];
    if ((t & 31) == 0) red[t >> 5] = s;
    __syncthreads();
    if (t == 0) {
        float a = 0.f;
        for (int i = 0; i < 8; ++i) a += red[i];
        cknorm[b] = fmaxf(sqrtf(a), 1e-6f);
    }
}

// ---------------------------------------------------------------------------
// K3: one wave per (b,n) row: similarity, occupancy, eff_protection, eff_usage.
// The single full-bandwidth pass over working_keys / working_values.
// ---------------------------------------------------------------------------
__global__ __launch_bounds__(256) void k_rowstats(
    const float* __restrict__ ck, const float* __restrict__ cknorm,
    const float* __restrict__ WK, const float* __restrict__ WV,
    const float* __restrict__ Woc, const float* __restrict__ boc,
    const float* __restrict__ Wpr, const float* __restrict__ bpr,
    const float* __restrict__ prot_in, const float* __restrict__ usage_in,
    float* __restrict__ simo, float* __restrict__ occo,
    float* __restrict__ epo, float* __restrict__ euo)
{
    const int row  = blockIdx.x * 8 + (threadIdx.x >> 5);   // b*N + n
    const int lane = threadIdx.x & 31;
    const int b    = row >> 7;                               // N = 128
    const float4* __restrict__ wk4 = (const float4*)(WK + (size_t)row * D_);
    const float4* __restrict__ wv4 = (const float4*)(WV + (size_t)row * D_);
    const float4* __restrict__ ck4 = (const float4*)(ck + (size_t)b * D_);
    const float4* __restrict__ oc4 = (const float4*)Woc;
    const float4* __restrict__ pr4 = (const float4*)Wpr;

    float s_ckwk = 0.f, s_wk2 = 0.f, s_wv2 = 0.f, s_oc = 0.f, s_pr = 0.f;
    #pragma unroll
    for (int i = 0; i < 4; ++i) {
        const int idx = i * 32 + lane;                       // coalesced float4
        const float4 wk = wk4[idx], wv = wv4[idx];
        const float4 c = ck4[idx], o = oc4[idx], p = pr4[idx];
        s_ckwk += wk.x * c.x + wk.y * c.y + wk.z * c.z + wk.w * c.w;
        s_wk2  += wk.x * wk.x + wk.y * wk.y + wk.z * wk.z + wk.w * wk.w;
        s_wv2  += wv.x * wv.x + wv.y * wv.y + wv.z * wv.z + wv.w * wv.w;
        s_oc   += wv.x * o.x + wv.y * o.y + wv.z * o.z + wv.w * o.w;
        s_pr   += wv.x * p.x + wv.y * p.y + wv.z * p.z + wv.w * p.w;
    }
    for (int off = 16; off; off >>= 1) {
        s_ckwk += __shfl_xor(s_ckwk, off);
        s_wk2  += __shfl_xor(s_wk2, off);
        s_wv2  += __shfl_xor(s_wv2, off);
        s_oc   += __shfl_xor(s_oc, off);
        s_pr   += __shfl_xor(s_pr, off);
    }
    if (lane == 0) {
        const float sim = s_ckwk / (cknorm[b] * fmaxf(sqrtf(s_wk2), 1e-6f));
        const float norm_occ = fminf(sqrtf(s_wv2) * 0.044194173824159216f /* 1/sqrt(512) */, 1.0f);
        const float occ = clamp01(0.5f * sigm(s_oc + boc[0]) + 0.5f * norm_occ);
        const float ep  = clamp01(0.4f * sigm(s_pr + bpr[0]) + 0.6f * prot_in[row]);
        const float eu  = clamp01(0.5f * occ + 0.5f * usage_in[row]);
        simo[row] = sim; occo[row] = occ; epo[row] = ep; euo[row] = eu;
    }
}

// ---------------------------------------------------------------------------
// K4: per-batch argmax / merge-vs-replace decision + overwrite & protection.
// softmax skipped: argmax(softmax(x/T)) == argmax(x), jnp tie-break = lowest idx.
// ---------------------------------------------------------------------------
__global__ __launch_bounds__(128) void k_decide(
    const float* __restrict__ sim, const float* __restrict__ occ,
    const float* __restrict__ eprot, const float* __restrict__ eusage,
    const float* __restrict__ age_in, const float* __restrict__ prot_in,
    const float* __restrict__ wstr, const float* __restrict__ imp,
    const float* __restrict__ mgl, const float* __restrict__ bdl,
    float* __restrict__ out_mp, float* __restrict__ out_bind, float* __restrict__ out_maxsim,
    float* __restrict__ out_ow, float* __restrict__ out_uprot,
    float* __restrict__ out_owmean, float* __restrict__ out_upmean,
    int* __restrict__ tgt_ws, float* __restrict__ um_ws,
    float* __restrict__ bind_ws, float* __restrict__ owt_ws)
{
    const int b = blockIdx.x, n = threadIdx.x;
    const int idx = b * N_ + n;
    const float s  = sim[idx];
    const float o  = occ[idx];
    const float ep = eprot[idx];
    const float eu = eusage[idx];
    const float ag = clamp01(age_in[idx]);
    const float rs = 1.15f * (1.f - o) + 0.85f * (1.f - ep) + 0.65f * ag
                   + 0.45f * (1.f - eu) + 0.25f * (1.f - s);

    __shared__ float sv[N_];
    __shared__ int   si[N_];
    __shared__ int   s_mi, s_ri, s_tgt;

    // argmax of similarity (merge)
    sv[n] = s; si[n] = n; __syncthreads();
    for (int st = 64; st; st >>= 1) {
        if (n < st) {
            const float v2 = sv[n + st]; const int i2 = si[n + st];
            if (v2 > sv[n] || (v2 == sv[n] && i2 < si[n])) { sv[n] = v2; si[n] = i2; }
        }
        __syncthreads();
    }
    if (n == 0) s_mi = si[0];
    __syncthreads();

    // argmax of replace score
    sv[n] = rs; si[n] = n; __syncthreads();
    for (int st = 64; st; st >>= 1) {
        if (n < st) {
            const float v2 = sv[n + st]; const int i2 = si[n + st];
            if (v2 > sv[n] || (v2 == sv[n] && i2 < si[n])) { sv[n] = v2; si[n] = i2; }
        }
        __syncthreads();
    }
    if (n == 0) s_ri = si[0];
    __syncthreads();

    if (n == 0) {
        const int mi = s_mi, ri = s_ri;
        const float ms   = sim[b * N_ + mi];
        const float mocc = occ[b * N_ + mi];
        const float mus  = eusage[b * N_ + mi];
        const float mage = clamp01(age_in[b * N_ + mi]);
        const float mp = sigm(mgl[b] + 2.4f * ms + 1.6f * (mocc - 0.5f)
                              + 1.0f * (mus - 0.5f) - 0.8f * mage);
        const bool cand = (ms > 0.55f) && (mocc > 0.35f);
        const bool um   = (mp >= 0.5f) && cand;
        const int  t    = um ? mi : ri;
        const float bv  = sigm(bdl[b] + 2.2f * ms);
        out_mp[b] = mp; out_bind[b] = bv; out_maxsim[b] = ms;
        tgt_ws[b] = t; um_ws[b] = um ? 1.f : 0.f; bind_ws[b] = bv;
        s_tgt = t;
    }
    __syncthreads();

    const int t = s_tgt;
    const float conflict = clamp01(1.f - s);
    const float ow = (n == t) ? (0.15f + 0.85f * wstr[b]) * (1.f - 0.65f * ep * conflict) : 0.f;
    out_ow[idx] = ow;
    if (n == t) owt_ws[b] = ow;
    const float boost = ow * (0.5f + 0.5f * imp[b]);
    const float up = clamp01(prot_in[idx] * 0.98f + boost);
    out_uprot[idx] = up;

    sv[n] = ow; __syncthreads();
    for (int st = 64; st; st >>= 1) { if (n < st) sv[n] += sv[n + st]; __syncthreads(); }
    if (n == 0) out_owmean[b] = sv[0] * (1.0f / N_);
    __syncthreads();
    sv[n] = up; __syncthreads();
    for (int st = 64; st; st >>= 1) { if (n < st) sv[n] += sv[n + st]; __syncthreads(); }
    if (n == 0) out_upmean[b] = sv[0] * (1.0f / N_);
}

// ---------------------------------------------------------------------------
// K5: rewrite only the target row of out_keys / out_values (overwrite == 0
// everywhere else, so the bulk memcpy already produced the correct output).
// ---------------------------------------------------------------------------
__global__ __launch_bounds__(128) void k_fixup(
    const float* __restrict__ WK, const float* __restrict__ WV,
    const float* __restrict__ ck, const float* __restrict__ cv,
    const int* __restrict__ tgt, const float* __restrict__ umf,
    const float* __restrict__ bind, const float* __restrict__ imp,
    const float* __restrict__ owt,
    float* __restrict__ out_keys, float* __restrict__ out_vals)
{
    const int b = blockIdx.x;
    const int t = tgt[b];
    const float ow = owt[b];
    const bool um = umf[b] > 0.5f;
    const float bs = bind[b], im = imp[b];
    const float kmix = um ? (0.22f + 0.38f * bs) : (0.78f + 0.18f * bs);
    const float vmix = um ? (0.45f + 0.35f * im) : (0.75f + 0.20f * im);
    const float ak = ow * kmix, av = ow * vmix;
    const size_t base = ((size_t)b * N_ + t) * D_;
    const float4* __restrict__ wk4 = (const float4*)(WK + base);
    const float4* __restrict__ wv4 = (const float4*)(WV + base);
    const float4* __restrict__ ck4 = (const float4*)(ck + (size_t)b * D_);
    const float4* __restrict__ cv4 = (const float4*)(cv + (size_t)b * D_);
    float4* __restrict__ ok4 = (float4*)(out_keys + base);
    float4* __restrict__ ov4 = (float4*)(out_vals + base);
    const int i = threadIdx.x;  // 128 float4s = 512 floats
    {
        const float4 wk = wk4[i], c = ck4[i];
        float4 r;
        r.x = wk.x + ak * (c.x - wk.x);
        r.y = wk.y + ak * (c.y - wk.y);
        r.z = wk.z + ak * (c.z - wk.z);
        r.w = wk.w + ak * (c.w - wk.w);
        ok4[i] = r;
    }
    {
        const float4 wv = wv4[i], c = cv4[i];
        float4 r;
        r.x = wv.x + av * (c.x - wv.x);
        r.y = wv.y + av * (c.y - wv.y);
        r.z = wv.z + av * (c.z - wv.z);
        r.w = wv.w + av * (c.w - wv.w);
        ov4[i] = r;
    }
}

// ---------------------------------------------------------------------------
extern "C" void kernel_launch(void* const* d_in, const int* in_sizes, int n_in,
                              void* d_out, int out_size, void* d_ws, size_t ws_size,
                              hipStream_t stream) {
    const float* hidden = (const float*)d_in[0];
    const float* wkeys  = (const float*)d_in[1];
    const float* wvals  = (const float*)d_in[2];
    const float* prot   = (const float*)d_in[3];
    const float* usage  = (const float*)d_in[4];
    const float* age    = (const float*)d_in[5];
    const float* Wk  = (const float*)d_in[6];
    const float* bk  = (const float*)d_in[7];
    const float* Wv  = (const float*)d_in[8];
    const float* bv  = (const float*)d_in[9];
    const float* Wwr = (const float*)d_in[10];
    const float* bwr = (const float*)d_in[11];
    const float* Wmg = (const float*)d_in[12];
    const float* bmg = (const float*)d_in[13];
    const float* Wbd = (const float*)d_in[14];
    const float* bbd = (const float*)d_in[15];
    const float* Wim = (const float*)d_in[16];
    const float* bim = (const float*)d_in[17];
    const float* Woc = (const float*)d_in[18];
    const float* boc = (const float*)d_in[19];
    const float* Wpr = (const float*)d_in[20];
    const float* bpr = (const float*)d_in[21];

    const size_t BND = (size_t)B_ * N_ * D_;
    const size_t BN  = (size_t)B_ * N_;
    const size_t BD  = (size_t)B_ * D_;

    float* out = (float*)d_out;
    float* out_keys   = out;
    float* out_vals   = out_keys + BND;
    float* out_uprot  = out_vals + BND;
    float* out_wstr   = out_uprot + BN;
    float* out_mp     = out_wstr + B_;
    float* out_bind   = out_mp + B_;
    float* out_owmean = out_bind + B_;
    float* out_upmean = out_owmean + B_;
    float* out_maxsim = out_upmean + B_;
    float* out_ow     = out_maxsim + B_;

    float* ws = (float*)d_ws;
    float* ck      = ws;            // B*D
    float* cv      = ck + BD;       // B*D
    float* wstr_ws = cv + BD;       // B
    float* imp_ws  = wstr_ws + B_;  // B
    float* mg_ws   = imp_ws + B_;   // B
    float* bd_ws   = mg_ws + B_;    // B
    float* cknorm  = bd_ws + B_;    // B
    float* um_ws   = cknorm + B_;   // B
    float* bind_ws = um_ws + B_;    // B
    float* owt_ws  = bind_ws + B_;  // B
    float* sim_ws  = owt_ws + B_;   // B*N
    float* occ_ws  = sim_ws + BN;   // B*N
    float* ep_ws   = occ_ws + BN;   // B*N
    float* eu_ws   = ep_ws + BN;    // B*N
    int*   tgt_ws  = (int*)(eu_ws + BN); // B

    // Bulk identity copies: updated_{keys,values} == working_{keys,values}
    // except at one row per batch, fixed up by k_fixup at the end.
    hipMemcpyAsync(out_keys, wkeys, BND * sizeof(float), hipMemcpyDeviceToDevice, stream);
    hipMemcpyAsync(out_vals, wvals, BND * sizeof(float), hipMemcpyDeviceToDevice, stream);

    k_rowproj<<<B_, 256, 0, stream>>>(hidden, Wwr, bwr, Wim, bim, Wmg, bmg, Wbd, bbd,
                                      wstr_ws, out_wstr, imp_ws, mg_ws, bd_ws);

    dim3 gemm_grid(B_ / 16, (D_ / 16) / 8, 2);
    k_gemm_tanh<<<gemm_grid, 256, 0, stream>>>(hidden, Wk, bk, Wv, bv, ck, cv);

    k_cknorm<<<B_, 256, 0, stream>>>(ck, cknorm);

    k_rowstats<<<(B_ * N_) / 8, 256, 0, stream>>>(ck, cknorm, wkeys, wvals,
                                                  Woc, boc, Wpr, bpr, prot, usage,
                                                  sim_ws, occ_ws, ep_ws, eu_ws);

    k_decide<<<B_, 128, 0, stream>>>(sim_ws, occ_ws, ep_ws, eu_ws, age, prot,
                                     wstr_ws, imp_ws, mg_ws, bd_ws,
                                     out_mp, out_bind, out_maxsim,
                                     out_ow, out_uprot, out_owmean, out_upmean,
                                     tgt_ws, um_ws, bind_ws, owt_ws);

    k_fixup<<<B_, 128, 0, stream>>>(wkeys, wvals, ck, cv,
                                    tgt_ws, um_ws, bind_ws, imp_ws, owt_ws,
                                    out_keys, out_vals);
}